// MultimodalGNN_13743895347695
// MI455X (gfx1250) — compile-verified
//
#include <hip/hip_runtime.h>
#include <stdint.h>

typedef float v2f __attribute__((ext_vector_type(2)));
typedef float v4f __attribute__((ext_vector_type(4)));
typedef float v8f __attribute__((ext_vector_type(8)));

#define NNODES 50000
#define SCAN_T 1024
#define CHUNK  128

// ---------------- degree / norm ----------------

__global__ void zero_u32(unsigned* __restrict__ p, int n) {
  int i = blockIdx.x * blockDim.x + threadIdx.x;
  if (i < n) p[i] = 0u;
}

__global__ void degree_count(const long long* __restrict__ dst, int E,
                             unsigned* __restrict__ deg) {
  int e = blockIdx.x * blockDim.x + threadIdx.x;
  if (e < E) atomicAdd(&deg[(int)dst[e]], 1u);
}

__global__ void deg_to_dinv(const unsigned* __restrict__ deg,
                            float* __restrict__ dinv, int n) {
  int i = blockIdx.x * blockDim.x + threadIdx.x;
  if (i < n) dinv[i] = rsqrtf((float)deg[i] + 1.0f);   // +1 = self loop
}

// Single-block exclusive scan over node degrees -> CSR row offsets.
// ~49 sequential elements per thread + 1024-wide Hillis-Steele scan in LDS.
__global__ void scan_offsets(const unsigned* __restrict__ deg, int n,
                             unsigned* __restrict__ row_start,
                             unsigned* __restrict__ cursor) {
  __shared__ unsigned part[SCAN_T];
  const int t = threadIdx.x;
  const int chunk = (n + SCAN_T - 1) / SCAN_T;
  const int beg = t * chunk;
  const int end = min(beg + chunk, n);
  unsigned s = 0;
  for (int i = beg; i < end; ++i) s += deg[i];
  part[t] = s;
  __syncthreads();
  for (int off = 1; off < SCAN_T; off <<= 1) {
    unsigned v = (t >= off) ? part[t - off] : 0u;
    __syncthreads();
    part[t] += v;
    __syncthreads();
  }
  unsigned run = part[t] - s;          // exclusive base for this thread
  for (int i = beg; i < end; ++i) {
    row_start[i] = run;
    cursor[i]    = run;
    run += deg[i];
  }
  if (t == SCAN_T - 1) row_start[n] = part[SCAN_T - 1];
}

// Bucket each edge into its destination row; per-edge weight computed once,
// reused by both layers.
__global__ void csr_fill(const long long* __restrict__ src,
                         const long long* __restrict__ dst,
                         const float* __restrict__ dinv,
                         unsigned* __restrict__ cursor,
                         int* __restrict__ csr_src,
                         float* __restrict__ csr_w, int E) {
  int e = blockIdx.x * blockDim.x + threadIdx.x;
  if (e >= E) return;
  int s = (int)src[e], d = (int)dst[e];
  unsigned pos = atomicAdd(&cursor[d], 1u);
  csr_src[pos] = s;
  csr_w[pos]   = dinv[s] * dinv[d];
}

// ---------------- fp32 WMMA GEMM: C[M,NF] = A[M,K] @ B[K,NF] ----------------
// One wave per 16(M) x 64(N) tile; 4 accumulators share one A fragment.
// blockDim.x = 32 * (NF/64), gridDim.x = M/16 (M = 50000 = 3125*16).
__global__ void wmma_gemm_f32(const float* __restrict__ A,
                              const float* __restrict__ B,
                              float* __restrict__ C,
                              int M, int K, int NF) {
  const int wave  = threadIdx.x >> 5;
  const int lane  = threadIdx.x & 31;
  const int r     = lane & 15;
  const int mbase = blockIdx.x << 4;
  const int nbase = wave << 6;
  const int koff  = (lane < 16) ? 0 : 2;

  const float* arow = A + (long)(mbase + r) * K;

  v8f c0 = {}; v8f c1 = {}; v8f c2 = {}; v8f c3 = {};

  for (int k = 0; k < K; k += 4) {
    v2f a = *(const v2f*)(arow + k + koff);           // K+koff, K+koff+1
    const float* b0 = B + (long)(k + koff) * NF + nbase + r;
    const float* b1 = b0 + NF;
    v2f bA = { b0[0],  b1[0]  };
    v2f bB = { b0[16], b1[16] };
    v2f bC = { b0[32], b1[32] };
    v2f bD = { b0[48], b1[48] };
    c0 = __builtin_amdgcn_wmma_f32_16x16x4_f32(false, a, false, bA, (short)0, c0, false, false);
    c1 = __builtin_amdgcn_wmma_f32_16x16x4_f32(false, a, false, bB, (short)0, c1, false, false);
    c2 = __builtin_amdgcn_wmma_f32_16x16x4_f32(false, a, false, bC, (short)0, c2, false, false);
    c3 = __builtin_amdgcn_wmma_f32_16x16x4_f32(false, a, false, bD, (short)0, c3, false, false);
  }

  const int mrow = mbase + ((lane < 16) ? 0 : 8);
  const int n    = nbase + r;
  #pragma unroll
  for (int j = 0; j < 8; ++j) {
    float* crow = C + (long)(mrow + j) * NF + n;
    crow[0]  = c0[j];
    crow[16] = c1[j];
    crow[32] = c2[j];
    crow[48] = c3[j];
  }
}

// ---------------- fused gather-aggregate + self-loop + bias (+ReLU) -----------
// One block per destination node, one thread per 4 features (float4 loads:
// 512B fetched per wave-instruction). Edge list staged through LDS and read
// as broadcasts; feature accesses fully coalesced. No atomics.
__global__ void gather_agg_v4(const float* __restrict__ h,
                              const int* __restrict__ csr_src,
                              const float* __restrict__ csr_w,
                              const unsigned* __restrict__ row_start,
                              const float* __restrict__ dinv,
                              const float* __restrict__ bias,
                              float* __restrict__ out,
                              int F, int do_relu) {
  __shared__ int   s_src[CHUNK];
  __shared__ float s_w[CHUNK];
  const int node = blockIdx.x;
  const int t    = threadIdx.x;       // 0 .. F/4-1
  const int f4   = t << 2;
  const int T    = blockDim.x;
  const unsigned beg = row_start[node];
  const unsigned end = row_start[node + 1];

  const float di = dinv[node];
  v4f acc = *(const v4f*)(h + (long)node * F + f4);
  acc *= di * di;                                     // self-loop message

  for (unsigned base = beg; base < end; base += CHUNK) {
    const int cnt = min((int)(end - base), CHUNK);
    __syncthreads();
    for (int j = t; j < cnt; j += T) {
      s_src[j] = csr_src[base + j];
      s_w[j]   = csr_w[base + j];
    }
    __syncthreads();
    for (int j = 0; j < cnt; ++j) {
      v4f hv = *(const v4f*)(h + (long)s_src[j] * F + f4);
      acc += hv * s_w[j];
    }
  }

  v4f v = acc + *(const v4f*)(bias + f4);
  if (do_relu) {
    #pragma unroll
    for (int i = 0; i < 4; ++i) v[i] = fmaxf(v[i], 0.0f);
  }
  *(v4f*)(out + (long)node * F + f4) = v;
}

// ---------------- launcher ----------------

static inline size_t align_elems(size_t n) { return (n + 63) & ~(size_t)63; } // 256B

extern "C" void kernel_launch(void* const* d_in, const int* in_sizes, int n_in,
                              void* d_out, int out_size, void* d_ws, size_t ws_size,
                              hipStream_t stream) {
  const float*     x    = (const float*)d_in[0];
  const long long* eidx = (const long long*)d_in[1];
  const float*     W1   = (const float*)d_in[2];
  const float*     b1   = (const float*)d_in[3];
  const float*     W2   = (const float*)d_in[4];
  const float*     b2   = (const float*)d_in[5];

  const int N  = NNODES;
  const int E  = in_sizes[1] / 2;
  const int F1 = 256, F2 = 128, Kd = 256;

  const long long* src = eidx;
  const long long* dst = eidx + E;

  // workspace layout (all 4-byte elems)
  unsigned* wsu = (unsigned*)d_ws;
  size_t o = 0;
  unsigned* deg       = wsu + o; o += align_elems((size_t)N);
  unsigned* row_start = wsu + o; o += align_elems((size_t)N + 1);
  unsigned* cursor    = wsu + o; o += align_elems((size_t)N);
  int*      csr_src   = (int*)(wsu + o);   o += align_elems((size_t)E);
  float*    csr_w     = (float*)(wsu + o); o += align_elems((size_t)E);
  float*    dinv      = (float*)(wsu + o); o += align_elems((size_t)N);
  float*    h1        = (float*)(wsu + o); o += align_elems((size_t)N * F1); // XW1, reused as HW2
  float*    agg1      = (float*)(wsu + o); o += align_elems((size_t)N * F1); // ReLU activations
  float*    out       = (float*)d_out;

  // --- degrees, dinv, CSR build (edge weights computed once, used twice) ---
  zero_u32<<<(N + 255) / 256, 256, 0, stream>>>(deg, N);
  degree_count<<<(E + 255) / 256, 256, 0, stream>>>(dst, E, deg);
  deg_to_dinv<<<(N + 255) / 256, 256, 0, stream>>>(deg, dinv, N);
  scan_offsets<<<1, SCAN_T, 0, stream>>>(deg, N, row_start, cursor);
  csr_fill<<<(E + 255) / 256, 256, 0, stream>>>(src, dst, dinv, cursor,
                                                csr_src, csr_w, E);

  // --- layer 1: h1 = x @ W1 (fp32 WMMA), then fused gather+bias+ReLU ---
  wmma_gemm_f32<<<N / 16, 32 * (F1 / 64), 0, stream>>>(x, W1, h1, N, Kd, F1);
  gather_agg_v4<<<N, F1 / 4, 0, stream>>>(h1, csr_src, csr_w, row_start, dinv,
                                          b1, agg1, F1, 1);

  // --- layer 2: h2 = act @ W2 (fp32 WMMA), then fused gather+bias into d_out ---
  float* h2 = h1;
  wmma_gemm_f32<<<N / 16, 32 * (F2 / 64), 0, stream>>>(agg1, W2, h2, N, Kd, F2);
  gather_agg_v4<<<N, F2 / 4, 0, stream>>>(h2, csr_src, csr_w, row_start, dinv,
                                          b2, out, F2, 0);
}